// LowRankCTRNN_71451075936494
// MI455X (gfx1250) — compile-verified
//
#include <hip/hip_runtime.h>
#include <hip/hip_bf16.h>

// ---------------------------------------------------------------------------
// LowRankCTRNN on MI455X (gfx1250, wave32)
//   Kernel 1: x_proj = input @ W_in^T + b_in   via V_WMMA_F32_16X16X4_F32
//             written directly into d_out's "out" region (in-place reuse).
//   Kernel 2: sequential scan over T, parallel over B (64 WGs x 2 batch rows),
//             h state in LDS, low-rank rec via per-wave dot + shfl_xor reduce.
// ---------------------------------------------------------------------------

typedef __attribute__((ext_vector_type(2))) float v2f;
typedef __attribute__((ext_vector_type(8))) float v8f;

namespace {
constexpr int kT = 512;
constexpr int kB = 128;
constexpr int kI = 64;
constexpr int kH = 512;
constexpr int kR = 8;
constexpr float kAlpha = 0.1f;
}

// ---------------------------------------------------------------------------
// Kernel 1: x_proj GEMM.  C[M=65536, N=512] = A[M,64] * B[64,N],
// A = input (row-major M x 64), B[k][n] = W_in[n][k] (W_in is H x I row-major).
// One wave computes one 16x16 tile: K=64 -> 16 chained WMMA f32 16x16x4.
// Grid: 4096 M-tiles * 32 N-tiles = 131072 waves = 16384 blocks of 8 waves.
// ---------------------------------------------------------------------------
__global__ __launch_bounds__(256) void
xproj_wmma_kernel(const float* __restrict__ inp,
                  const float* __restrict__ W_in,
                  const float* __restrict__ b_in,
                  float* __restrict__ xp)
{
    const int lane    = threadIdx.x & 31;
    const int waveBlk = threadIdx.x >> 5;
    const int w       = blockIdx.x * 8 + waveBlk;   // global wave id
    const int mt      = w >> 5;                     // 0..4095
    const int nt      = w & 31;                     // 0..31
    const int halfSel = lane >> 4;                  // 0 | 1
    const int lane16  = lane & 15;

    // ISA 7.12.2, 32-bit A 16x4: lanes 0-15 hold K=k0,k0+1; lanes 16-31 K=k0+2,k0+3.
    // B 4x16 mirrored: lane gives N, halfSel gives K pair.
    const int mRow = mt * 16 + lane16;
    const int nCol = nt * 16 + lane16;

    const float* __restrict__ Arow = inp  + (size_t)mRow * kI;
    const float* __restrict__ Brow = W_in + (size_t)nCol * kI;   // W_in[n][*]

    v8f c = {};
#pragma unroll
    for (int k0 = 0; k0 < kI; k0 += 4) {
        const int kk = k0 + halfSel * 2;
        v2f a = *(const v2f*)(Arow + kk);
        v2f b = *(const v2f*)(Brow + kk);
        // (neg_a, A, neg_b, B, c_mod, C, reuse_a, reuse_b)
        c = __builtin_amdgcn_wmma_f32_16x16x4_f32(
                false, a, false, b, (short)0, c, false, false);
    }

    const float bv = b_in[nCol];
    // D layout: VGPR j -> row (j + halfSel*8), col = lane16.
#pragma unroll
    for (int j = 0; j < 8; ++j) {
        const int row = mt * 16 + halfSel * 8 + j;
        xp[(size_t)row * kH + nCol] = c[j] + bv;
    }
}

// ---------------------------------------------------------------------------
// Kernel 2: scan.  64 blocks x 512 threads; block handles batch rows
// {2*blk, 2*blk+1} for all 512 steps.  h[2][512] in LDS.
//   wave w (0..15) owns (b = w>>3, r = w&7): p[b][r] = dot(h[b], V[r]) over H.
//   thread j owns h element j for both rows in the elementwise phase.
// x_proj is read from `out` and overwritten in place with h each step.
// ---------------------------------------------------------------------------
__global__ __launch_bounds__(512) void
ctrnn_scan_kernel(const float* __restrict__ noise,
                  const float* __restrict__ U,   // (H, R) row-major
                  const float* __restrict__ V,   // (R, H) row-major
                  float* __restrict__ out)       // (T, B, H) then hN (B, H)
{
    __shared__ float h_s[2 * kH];
    __shared__ float p_s[16];

    const int j    = threadIdx.x;        // 0..511  -> H index
    const int lane = j & 31;
    const int wv   = j >> 5;             // 0..15
    const int myB  = wv >> 3;            // 0 | 1
    const int myR  = wv & 7;             // 0..7
    const int b0   = blockIdx.x * 2;     // first batch row of this block

    // Per-thread constants.
    float Ureg[kR];
#pragma unroll
    for (int r = 0; r < kR; ++r) Ureg[r] = U[(size_t)j * kR + r];

    float Vreg[16];                      // V[myR][lane + 32k], k = 0..15
#pragma unroll
    for (int k = 0; k < 16; ++k) Vreg[k] = V[(size_t)myR * kH + lane + 32 * k];

    h_s[j]      = 0.0f;                  // h0 = 0
    h_s[kH + j] = 0.0f;
    __syncthreads();

    // Software-pipelined load stream: regs hold step t, loads for t+1 issue
    // at the top of iteration t (store to out[t] happens after, no conflict).
    size_t i0 = ((size_t)0 * kB + b0) * kH + j;
    size_t i1 = i0 + kH;
    float xp0 = out[i0],   xp1 = out[i1];
    float nz0 = noise[i0], nz1 = noise[i1];

    for (int t = 0; t < kT; ++t) {
        float xp0n = 0.f, xp1n = 0.f, nz0n = 0.f, nz1n = 0.f;
        size_t n0 = 0, n1 = 0;
        if (t + 1 < kT) {
            n0 = ((size_t)(t + 1) * kB + b0) * kH + j;
            n1 = n0 + kH;
            xp0n = out[n0];   xp1n = out[n1];
            nz0n = noise[n0]; nz1n = noise[n1];
        }
        if (t + 2 < kT) {   // L2 prefetch two steps ahead (global_prefetch_b8)
            size_t pf = ((size_t)(t + 2) * kB + b0) * kH + j;
            __builtin_prefetch(&out[pf], 0, 1);
            __builtin_prefetch(&noise[pf], 0, 1);
        }

        // h += noise  (thread j owns h element j of both rows)
        float h0 = h_s[j]      + nz0;
        float h1 = h_s[kH + j] + nz1;
        h_s[j]      = h0;
        h_s[kH + j] = h1;
        __syncthreads();

        // p[b][r] = dot(h[b], V[r]) : one wave per (b,r)
        float part = 0.0f;
#pragma unroll
        for (int k = 0; k < 16; ++k)
            part = fmaf(h_s[myB * kH + lane + 32 * k], Vreg[k], part);
#pragma unroll
        for (int off = 16; off >= 1; off >>= 1)
            part += __shfl_xor(part, off);
        if (lane == 0) p_s[wv] = part;
        __syncthreads();

        // Elementwise update + store (overwrites x_proj slot with h).
        float rec0 = 0.0f, rec1 = 0.0f;
#pragma unroll
        for (int r = 0; r < kR; ++r) {
            rec0 = fmaf(p_s[r],     Ureg[r], rec0);
            rec1 = fmaf(p_s[8 + r], Ureg[r], rec1);
        }
        float hn0 = fmaxf(xp0 + rec0, 0.0f);
        float hn1 = fmaxf(xp1 + rec1, 0.0f);
        h0 = h0 * (1.0f - kAlpha) + hn0 * kAlpha;
        h1 = h1 * (1.0f - kAlpha) + hn1 * kAlpha;
        h_s[j]      = h0;
        h_s[kH + j] = h1;

        size_t o0 = ((size_t)t * kB + b0) * kH + j;
        out[o0]      = h0;
        out[o0 + kH] = h1;

        // rotate pipeline regs
        xp0 = xp0n; xp1 = xp1n; nz0 = nz0n; nz1 = nz1n;
        __syncthreads();   // h_s writes visible before next reduction
    }

    // hN region starts after out: offset T*B*H.  hN == final h.
    const size_t hN = (size_t)kT * kB * kH;
    out[hN + (size_t)(b0 + 0) * kH + j] = h_s[j];
    out[hN + (size_t)(b0 + 1) * kH + j] = h_s[kH + j];
}

// ---------------------------------------------------------------------------
extern "C" void kernel_launch(void* const* d_in, const int* in_sizes, int n_in,
                              void* d_out, int out_size, void* d_ws, size_t ws_size,
                              hipStream_t stream) {
    (void)in_sizes; (void)n_in; (void)out_size; (void)d_ws; (void)ws_size;

    const float* inp   = (const float*)d_in[0];  // (T, B, I)
    const float* W_in  = (const float*)d_in[1];  // (H, I)
    const float* b_in  = (const float*)d_in[2];  // (H,)
    const float* U     = (const float*)d_in[3];  // (H, R)
    const float* V     = (const float*)d_in[4];  // (R, H)
    const float* noise = (const float*)d_in[5];  // (T, B, H)
    float* out = (float*)d_out;                  // (T,B,H) out ++ (B,H) hN

    // Kernel 1: 4096 M-tiles * 32 N-tiles, 8 waves/block -> 16384 blocks.
    xproj_wmma_kernel<<<16384, 256, 0, stream>>>(inp, W_in, b_in, out);

    // Kernel 2: 64 blocks (2 batch rows each) x 512 threads.
    ctrnn_scan_kernel<<<64, 512, 0, stream>>>(noise, U, V, out);
}